// surface_loss_6863357739233
// MI455X (gfx1250) — compile-verified
//
#include <hip/hip_runtime.h>

#define IMG 128
#define NPIX (IMG * IMG)   // 16384
#define NB 64              // batch
#define SAMPLE 1024
#define NV 778
#define NVPAD 784          // 49 * 16
#define HCNT 524288u       // half counter for 64*16384 = 1048576 threefry elements

typedef float v2f __attribute__((ext_vector_type(2)));
typedef float v8f __attribute__((ext_vector_type(8)));

// ---------------------------------------------------------------------------
// threefry2x32 with key = jax.random.key(1) -> (k0,k1) = (0,1)
// ---------------------------------------------------------------------------
__device__ __forceinline__ unsigned rotl32(unsigned x, unsigned d) {
  return (x << d) | (x >> (32u - d));
}

__device__ __forceinline__ void threefry2x32(unsigned x0, unsigned x1,
                                             unsigned& o0, unsigned& o1) {
  const unsigned k0 = 0u, k1 = 1u;
  const unsigned k2 = k0 ^ k1 ^ 0x1BD11BDAu;
  x0 += k0; x1 += k1;
#define TF_R(r) { x0 += x1; x1 = rotl32(x1, r); x1 ^= x0; }
  TF_R(13u) TF_R(15u) TF_R(26u) TF_R(6u)
  x0 += k1; x1 += k2 + 1u;
  TF_R(17u) TF_R(29u) TF_R(16u) TF_R(24u)
  x0 += k2; x1 += k0 + 2u;
  TF_R(13u) TF_R(15u) TF_R(26u) TF_R(6u)
  x0 += k0; x1 += k1 + 3u;
  TF_R(17u) TF_R(29u) TF_R(16u) TF_R(24u)
  x0 += k1; x1 += k2 + 4u;
  TF_R(13u) TF_R(15u) TF_R(26u) TF_R(6u)
  x0 += k2; x1 += k0 + 5u;
#undef TF_R
  o0 = x0; o1 = x1;
}

// Orderable u32 key from float (monotone: a<b  <=>  key(a)<key(b))
__device__ __forceinline__ unsigned ord_key(float f) {
  unsigned b = __float_as_uint(f);
  return (b & 0x80000000u) ? ~b : (b | 0x80000000u);
}

// Gumbel score -> orderable key, exactly mirroring the reference RNG path.
__device__ unsigned score_key(const float* __restrict__ real, int b, int p) {
  unsigned n = (unsigned)(b * NPIX + p);
  unsigned o0, o1, bits;
  if (n < HCNT) { threefry2x32(n, n + HCNT, o0, o1); bits = o0; }
  else          { threefry2x32(n - HCNT, n, o0, o1); bits = o1; }
  float u = __uint_as_float((bits >> 9) | 0x3F800000u) - 1.0f;  // [0,1)
  float g = -logf(-logf(u + 1e-20f) + 1e-20f);
  float img = real[b * NPIX + p];
  float sc = (img <= 0.99f) ? g : -__builtin_inff();
  return ord_key(sc);
}

// ---------------------------------------------------------------------------
// K1: per-batch top-1024 radix select (1 block / batch, 1024 threads,
// 16 pixels per thread kept in registers). Deterministic slot assignment via
// block prefix scan; ties at threshold take lowest pixel index (XLA TopK).
// ---------------------------------------------------------------------------
__global__ __launch_bounds__(1024) void select_kernel(const float* __restrict__ real,
                                                      unsigned* __restrict__ idx_out) {
  const int b = blockIdx.x;
  const int t = threadIdx.x;
  unsigned keys[16];
  const int p0 = t * 16;
#pragma unroll
  for (int i = 0; i < 16; ++i) keys[i] = score_key(real, b, p0 + i);

  __shared__ unsigned hist[256];
  __shared__ unsigned sSel, sAbove;
  __shared__ unsigned scan[1024];

  unsigned prefix = 0u;
  unsigned k = SAMPLE;  // remaining rank within prefix-matched keys
  for (int pass = 0; pass < 4; ++pass) {
    const int shift = 24 - 8 * pass;
    const unsigned him = (pass == 0) ? 0u : (0xFFFFFFFFu << (shift + 8));
    if (t < 256) hist[t] = 0u;
    __syncthreads();
#pragma unroll
    for (int i = 0; i < 16; ++i) {
      unsigned kk = keys[i];
      if ((kk & him) == prefix) atomicAdd(&hist[(kk >> shift) & 0xFFu], 1u);
    }
    __syncthreads();
    if (t == 0) {
      unsigned cum = 0u, sel = 0u;
      for (int bkt = 255; bkt >= 0; --bkt) {
        unsigned h = hist[bkt];
        if (cum + h >= k) { sel = (unsigned)bkt; break; }
        cum += h;
      }
      sSel = sel; sAbove = cum;
    }
    __syncthreads();
    prefix |= (sSel << shift);
    k -= sAbove;
    __syncthreads();
  }
  const unsigned T = prefix;          // threshold key (1024-th largest)
  const unsigned G = SAMPLE - k;      // count of keys strictly greater than T

  unsigned cgt = 0u, ceq = 0u;
#pragma unroll
  for (int i = 0; i < 16; ++i) {
    if (keys[i] > T) ++cgt; else if (keys[i] == T) ++ceq;
  }
  unsigned packed = cgt | (ceq << 16);
  scan[t] = packed;
  __syncthreads();
  for (int off = 1; off < 1024; off <<= 1) {
    unsigned v = scan[t];
    unsigned a = (t >= off) ? scan[t - off] : 0u;
    __syncthreads();
    scan[t] = v + a;
    __syncthreads();
  }
  const unsigned excl = scan[t] - packed;
  unsigned go = excl & 0xFFFFu;   // rank among strictly-greater (pixel order)
  unsigned eq = excl >> 16;       // rank among equal-to-threshold
#pragma unroll
  for (int i = 0; i < 16; ++i) {
    unsigned kk = keys[i];
    if (kk > T) {
      idx_out[b * SAMPLE + go] = (unsigned)(p0 + i); ++go;
    } else if (kk == T) {
      if (eq < k) idx_out[b * SAMPLE + G + eq] = (unsigned)(p0 + i);
      ++eq;
    }
  }
}

// ---------------------------------------------------------------------------
// K2: gather sampled pixels and run img2pcl; store padded (x,y,z,0) A-rows
// and |p|^2.
// ---------------------------------------------------------------------------
__global__ __launch_bounds__(256) void gather_kernel(const float* __restrict__ real,
                                                     const float* __restrict__ center,
                                                     const float* __restrict__ Mmat,
                                                     const float* __restrict__ cube,
                                                     const unsigned* __restrict__ idx,
                                                     float* __restrict__ pclA,
                                                     float* __restrict__ pn) {
  const int b = blockIdx.y;
  const int s = blockIdx.x * blockDim.x + threadIdx.x;
  if (s >= SAMPLE) return;
  const unsigned p = idx[b * SAMPLE + s];
  const int ii = (int)(p >> 7);
  const int jj = (int)(p & 127u);
  const float img = real[b * NPIX + (int)p];

  const float* Mb = Mmat + b * 9;
  const float a00 = Mb[0], a01 = Mb[1], a02 = Mb[2];
  const float a10 = Mb[3], a11 = Mb[4], a12 = Mb[5];
  const float a20 = Mb[6], a21 = Mb[7], a22 = Mb[8];
  const float det = a00 * (a11 * a22 - a12 * a21)
                  - a01 * (a10 * a22 - a12 * a20)
                  + a02 * (a10 * a21 - a11 * a20);
  const float rd = 1.0f / det;
  const float i00 =  (a11 * a22 - a12 * a21) * rd;
  const float i01 = -(a01 * a22 - a02 * a21) * rd;
  const float i02 =  (a01 * a12 - a02 * a11) * rd;
  const float i10 = -(a10 * a22 - a12 * a20) * rd;
  const float i11 =  (a00 * a22 - a02 * a20) * rd;
  const float i12 = -(a00 * a12 - a02 * a10) * rd;

  const float u0 = (float)jj * (128.0f / 127.0f);  // (t_j+1)*IMG/2
  const float u1 = (float)ii * (128.0f / 127.0f);
  const float w0 = i00 * u0 + i01 * u1 + i02;
  const float w1 = i10 * u0 + i11 * u1 + i12;

  const float cx = center[b * 3 + 0], cy = center[b * 3 + 1], cz = center[b * 3 + 2];
  const float hx = cube[b * 3 + 0] * 0.5f, hy = cube[b * 3 + 1] * 0.5f, hz = cube[b * 3 + 2] * 0.5f;
  const float d = img * hz + cz;
  const float x = (w0 - 320.0f) * d / 588.03f;
  const float y = (w1 - 240.0f) * d / 587.07f;   // FLIP = 1.0
  const float xn = (x - cx) / hx;
  const float yn = (y - cy) / hy;
  const float zn = (d - cz) / hz;

  float* o = pclA + ((size_t)(b * SAMPLE + s)) * 4;
  o[0] = xn; o[1] = yn; o[2] = zn; o[3] = 0.0f;
  pn[b * SAMPLE + s] = xn * xn + yn * yn + zn * zn;
}

// ---------------------------------------------------------------------------
// K3: Chamfer via V_WMMA_F32_16X16X4_F32. One wave per (batch, 16-point tile).
// d2 = |p|^2 + |v|^2 - 2 p.v ; row mins in registers, col mins via
// deterministic u32 atomicMin on orderable keys.
// ---------------------------------------------------------------------------
__global__ __launch_bounds__(32) void chamfer_kernel(const float* __restrict__ pclA,
                                                     const float* __restrict__ pn,
                                                     const float* __restrict__ verts,
                                                     float* __restrict__ minv,
                                                     unsigned* __restrict__ minpK) {
  const int b = blockIdx.y;
  const int rt = blockIdx.x;
  const int lane = threadIdx.x;
  const int l15 = lane & 15;
  const bool hi = lane >= 16;
  const int rowbase = rt * 16;

  // A tile 16x4 (M x K): lanes 0-15 -> (K0,K1)=(x,y); lanes 16-31 -> (K2,K3)=(z,0)
  const float* ap = pclA + ((size_t)(b * SAMPLE + rowbase + l15)) * 4 + (hi ? 2 : 0);
  v2f A; A.x = ap[0]; A.y = ap[1];

  const float pnl = pn[b * SAMPLE + rowbase + l15];
  float prow[8];
#pragma unroll
  for (int r = 0; r < 8; ++r) prow[r] = __shfl(pnl, (hi ? 8 : 0) + r, 32);

  float rowmin[8];
#pragma unroll
  for (int r = 0; r < 8; ++r) rowmin[r] = __builtin_inff();

  for (int ct = 0; ct < NVPAD / 16; ++ct) {
    const int c = ct * 16 + l15;
    // B tile 4x16 (K x N): lanes 0-15 -> (K0,K1)=(x,y); lanes 16-31 -> (K2,K3)=(z,0)
    v2f Bv;
    if (c < NV) {
      const float* vp = verts + ((size_t)(b * NV + c)) * 3;
      if (!hi) { Bv.x = vp[0]; Bv.y = vp[1]; }
      else     { Bv.x = vp[2]; Bv.y = 0.0f; }
    } else {
      Bv.x = 1.0e9f; Bv.y = hi ? 0.0f : 1.0e9f;  // huge finite pad -> never min
    }
    float sq = Bv.x * Bv.x + Bv.y * Bv.y;
    float vn = sq + __shfl_xor(sq, 16, 32);  // full |v|^2 on both half-lanes

    v8f Cz = {0.f, 0.f, 0.f, 0.f, 0.f, 0.f, 0.f, 0.f};
    v8f D = __builtin_amdgcn_wmma_f32_16x16x4_f32(
        false, A, false, Bv, (short)0, Cz, false, false);

    float cmin = __builtin_inff();
#pragma unroll
    for (int r = 0; r < 8; ++r) {
      float d2 = prow[r] + vn - 2.0f * D[r];
      rowmin[r] = fminf(rowmin[r], d2);
      cmin = fminf(cmin, d2);
    }
    cmin = fminf(cmin, __shfl_xor(cmin, 16, 32));
    if (!hi && c < NV) atomicMin(&minpK[b * NV + c], ord_key(cmin));
  }
#pragma unroll
  for (int off = 1; off < 16; off <<= 1) {
#pragma unroll
    for (int r = 0; r < 8; ++r) rowmin[r] = fminf(rowmin[r], __shfl_xor(rowmin[r], off, 32));
  }
  if (l15 == 0) {
#pragma unroll
    for (int r = 0; r < 8; ++r)
      minv[(size_t)b * SAMPLE + rowbase + (hi ? 8 : 0) + r] = rowmin[r];
  }
}

__global__ void init_minp_kernel(unsigned* __restrict__ minpK, int n) {
  int i = blockIdx.x * blockDim.x + threadIdx.x;
  if (i < n) minpK[i] = 0xFFFFFFFFu;  // acts as +inf in orderable-key space
}

__global__ __launch_bounds__(256) void finalize_kernel(const float* __restrict__ minv,
                                                       const unsigned* __restrict__ minpK,
                                                       float* __restrict__ out) {
  __shared__ float red[256];
  const int t = threadIdx.x;
  float s1 = 0.0f;
  for (int i = t; i < NB * SAMPLE; i += 256) s1 += minv[i];
  float s2 = 0.0f;
  for (int i = t; i < NB * NV; i += 256) {
    unsigned kk = minpK[i];
    unsigned bits = (kk & 0x80000000u) ? (kk & 0x7FFFFFFFu) : ~kk;  // invert ord_key
    s2 += __uint_as_float(bits);
  }
  red[t] = s1; __syncthreads();
  for (int off = 128; off > 0; off >>= 1) { if (t < off) red[t] += red[t + off]; __syncthreads(); }
  float S1 = red[0]; __syncthreads();
  red[t] = s2; __syncthreads();
  for (int off = 128; off > 0; off >>= 1) { if (t < off) red[t] += red[t + off]; __syncthreads(); }
  if (t == 0) out[0] = S1 / (float)(NB * SAMPLE) + red[0] / (float)(NB * NV);
}

// ---------------------------------------------------------------------------
extern "C" void kernel_launch(void* const* d_in, const int* in_sizes, int n_in,
                              void* d_out, int out_size, void* d_ws, size_t ws_size,
                              hipStream_t stream) {
  (void)in_sizes; (void)n_in; (void)out_size; (void)ws_size;
  const float* real   = (const float*)d_in[0];
  // d_in[1] = synth (unused by the loss), d_in[3] = faces (unused)
  const float* verts  = (const float*)d_in[2];
  const float* center = (const float*)d_in[4];
  const float* Mmat   = (const float*)d_in[5];
  const float* cube   = (const float*)d_in[6];

  // Workspace layout (~2.0 MB total)
  unsigned* idx   = (unsigned*)d_ws;                       // NB*SAMPLE u32
  float* pclA     = (float*)(idx + NB * SAMPLE);           // NB*SAMPLE*4 f32
  float* pn       = pclA + (size_t)NB * SAMPLE * 4;        // NB*SAMPLE f32
  float* minv     = pn + (size_t)NB * SAMPLE;              // NB*SAMPLE f32
  unsigned* minpK = (unsigned*)(minv + (size_t)NB * SAMPLE); // NB*NV u32

  init_minp_kernel<<<(NB * NV + 255) / 256, 256, 0, stream>>>(minpK, NB * NV);
  select_kernel<<<NB, 1024, 0, stream>>>(real, idx);
  gather_kernel<<<dim3(SAMPLE / 256, NB), 256, 0, stream>>>(real, center, Mmat, cube,
                                                            idx, pclA, pn);
  chamfer_kernel<<<dim3(SAMPLE / 16, NB), 32, 0, stream>>>(pclA, pn, verts, minv, minpK);
  finalize_kernel<<<1, 256, 0, stream>>>(minv, minpK, (float*)d_out);
}